// EmilyGIN_bond_87703232184759
// MI455X (gfx1250) — compile-verified
//
#include <hip/hip_runtime.h>

#define D 128
#define BN_EPS 1e-5f
#define NCHUNKS 400

typedef float v2f __attribute__((ext_vector_type(2)));
typedef float v8f __attribute__((ext_vector_type(8)));

// ---------------------------------------------------------------- init: h = feature
__global__ void gin_init_h(const float4* __restrict__ feat, float4* __restrict__ h, int n4) {
    int i = blockIdx.x * blockDim.x + threadIdx.x;
    if (i < n4) h[i] = feat[i];
}

// ---------------------------------------------------------------- scatter: h[dst] += feature[src]
// one wave per edge, each lane handles 4 consecutive floats (32*4 = 128 = D)
__global__ void gin_scatter(const float* __restrict__ feat, const int* __restrict__ ei,
                            float* __restrict__ h, int n_edges) {
    int t = blockIdx.x * blockDim.x + threadIdx.x;
    int e = t >> 5;
    int lane = t & 31;
    if (e >= n_edges) return;
    int s = ei[e];             // edge_index[0][e] (src)
    int d = ei[n_edges + e];   // edge_index[1][e] (dst)
    const float4 v = *(const float4*)(feat + (size_t)s * D + lane * 4);
    float* dp = h + (size_t)d * D + lane * 4;
    atomicAdd(dp + 0, v.x);
    atomicAdd(dp + 1, v.y);
    atomicAdd(dp + 2, v.z);
    atomicAdd(dp + 3, v.w);
}

// ---------------------------------------------------------------- out = relu(in @ W.T + bias)
// in: n x 128 row-major, W: 128 x 128 row-major (torch Linear weight)
// one wave computes one 16x16 tile; 8 waves/block cover all 8 column tiles (D/16).
// K-loop: 32 chained V_WMMA_F32_16X16X4_F32 accumulations (full f32 precision).
__global__ void gin_gemm_relu(const float* __restrict__ in, const float* __restrict__ W,
                              const float* __restrict__ bias, float* __restrict__ out) {
    const int lane  = threadIdx.x & 31;
    const int wv    = threadIdx.x >> 5;       // 0..7 : column tile index
    const int m     = lane & 15;              // A row / B col / C col within tile
    const int koff  = (lane >> 4) * 2;        // K sub-offset per ISA 16x4 f32 A layout
    const int itile = blockIdx.x * 16;
    const int jtile = wv * 16;

    // A fragment: A(m, k0 + koff .. +1)      -> contiguous float2 in row m of `in`
    // B fragment: B(k, n) = W.T[k][n] = W[n][k] -> contiguous float2 in row n of `W`
    const float* arow = in + (size_t)(itile + m) * D + koff;
    const float* brow = W  + (size_t)(jtile + m) * D + koff;

    v8f c = {};
#pragma unroll
    for (int k = 0; k < D; k += 4) {
        v2f a = *(const v2f*)(arow + k);
        v2f b = *(const v2f*)(brow + k);
        c = __builtin_amdgcn_wmma_f32_16x16x4_f32(false, a, false, b, (short)0, c,
                                                  false, false);
    }

    // C/D layout: VGPR v -> row (v + 8*(lane>>4)), col (lane&15)
    const float bb = bias[jtile + m];
    const int hi = (lane >> 4) * 8;
#pragma unroll
    for (int v = 0; v < 8; ++v) {
        out[(size_t)(itile + v + hi) * D + jtile + m] = fmaxf(c[v] + bb, 0.0f);
    }
}

// ---------------------------------------------------------------- BN stage 1: per-chunk column sums
__global__ void gin_col_partial(const float* __restrict__ h, float* __restrict__ ps,
                                float* __restrict__ ps2, int n_nodes, int rows_per) {
    int col = threadIdx.x;                    // 0..127
    int r0 = blockIdx.x * rows_per;
    int r1 = r0 + rows_per;
    if (r1 > n_nodes) r1 = n_nodes;
    float s = 0.0f, s2 = 0.0f;
    for (int r = r0; r < r1; ++r) {
        float v = h[(size_t)r * D + col];
        s += v;
        s2 += v * v;
    }
    ps [blockIdx.x * D + col] = s;
    ps2[blockIdx.x * D + col] = s2;
}

// ---------------------------------------------------------------- BN stage 2: mean + rsqrt(var+eps)
__global__ void gin_bn_stats(const float* __restrict__ ps, const float* __restrict__ ps2,
                             float* __restrict__ mstd, int nchunks, int n_nodes) {
    int col = threadIdx.x;
    float s = 0.0f, s2 = 0.0f;
    for (int c = 0; c < nchunks; ++c) {
        s  += ps [c * D + col];
        s2 += ps2[c * D + col];
    }
    float inv_n = 1.0f / (float)n_nodes;
    float mean = s * inv_n;
    float var = fmaxf(s2 * inv_n - mean * mean, 0.0f);
    mstd[col]     = mean;
    mstd[D + col] = rsqrtf(var + BN_EPS);
}

// ---------------------------------------------------------------- BN apply
__global__ void gin_bn_apply(const float4* __restrict__ h, const float* __restrict__ mstd,
                             const float* __restrict__ gamma, const float* __restrict__ beta,
                             float4* __restrict__ out, int n4) {
    int i = blockIdx.x * blockDim.x + threadIdx.x;
    if (i >= n4) return;
    int c = (i & 31) * 4;                     // 32 float4 per row of 128
    float4 v  = h[i];
    float4 mn = *(const float4*)(mstd + c);
    float4 rs = *(const float4*)(mstd + D + c);
    float4 g  = *(const float4*)(gamma + c);
    float4 b  = *(const float4*)(beta + c);
    float4 o;
    o.x = (v.x - mn.x) * rs.x * g.x + b.x;
    o.y = (v.y - mn.y) * rs.y * g.y + b.y;
    o.z = (v.z - mn.z) * rs.z * g.z + b.z;
    o.w = (v.w - mn.w) * rs.w * g.w + b.w;
    out[i] = o;
}

extern "C" void kernel_launch(void* const* d_in, const int* in_sizes, int n_in,
                              void* d_out, int out_size, void* d_ws, size_t ws_size,
                              hipStream_t stream) {
    const float* feat  = (const float*)d_in[0];
    const int*   ei    = (const int*)d_in[1];
    const float* w1    = (const float*)d_in[2];
    const float* b1    = (const float*)d_in[3];
    const float* w2    = (const float*)d_in[4];
    const float* b2    = (const float*)d_in[5];
    const float* gamma = (const float*)d_in[6];
    const float* beta  = (const float*)d_in[7];
    float* out = (float*)d_out;

    const int n_nodes = in_sizes[0] / D;      // 100000
    const int n_edges = in_sizes[1] / 2;      // 400000
    const int n4 = n_nodes * (D / 4);         // float4 count

    // ws layout: [ h buffer: n_nodes*D floats | ps: NCHUNKS*D | ps2: NCHUNKS*D | mstd: 2*D ]
    float* hbuf = (float*)d_ws;
    float* ps   = hbuf + (size_t)n_nodes * D;
    float* ps2  = ps  + NCHUNKS * D;
    float* mstd = ps2 + NCHUNKS * D;

    // 1. h = feature
    gin_init_h<<<(n4 + 255) / 256, 256, 0, stream>>>((const float4*)feat, (float4*)hbuf, n4);
    // 2. h[dst] += feature[src]  (one wave per edge)
    long long nthr = (long long)n_edges * 32;
    gin_scatter<<<(int)((nthr + 255) / 256), 256, 0, stream>>>(feat, ei, hbuf, n_edges);
    // 3. h1 = relu(h @ w1.T + b1)  -> stored in d_out (ping-pong to keep ws small)
    gin_gemm_relu<<<n_nodes / 16, 256, 0, stream>>>(hbuf, w1, b1, out);
    // 4. h2 = relu(h1 @ w2.T + b2) -> back into ws (overwrites h, no longer needed)
    gin_gemm_relu<<<n_nodes / 16, 256, 0, stream>>>(out, w2, b2, hbuf);
    // 5-6. batch-norm stats (deterministic two-stage reduction)
    int rows_per = (n_nodes + NCHUNKS - 1) / NCHUNKS;
    gin_col_partial<<<NCHUNKS, 128, 0, stream>>>(hbuf, ps, ps2, n_nodes, rows_per);
    gin_bn_stats<<<1, 128, 0, stream>>>(ps, ps2, mstd, NCHUNKS, n_nodes);
    // 7. normalize -> d_out
    gin_bn_apply<<<(n4 + 255) / 256, 256, 0, stream>>>((const float4*)hbuf, mstd, gamma, beta,
                                                       (float4*)out, n4);
}